// SkipGramNS_33457795235918
// MI455X (gfx1250) — compile-verified
//
#include <hip/hip_runtime.h>

typedef float v2f __attribute__((ext_vector_type(2)));
typedef float v8f __attribute__((ext_vector_type(8)));

// -log_sigmoid(x) = softplus(-x);  softplus(x) = max(x,0) + log(1 + exp(-|x|))
__device__ __forceinline__ float softplus_fast(float x) {
    return fmaxf(x, 0.0f) + __logf(1.0f + __expf(-fabsf(x)));
}

// One wave32 per sample b.
// WMMA f32 16x16x4, accumulated over D/4 chunks:
//   A (16x4): rows = context vectors (row0=positive, rows1..K=negatives,
//             rows K+1..15 aliased to positive; ignored at readout)
//   B (4x16): target vector chunk broadcast to all 16 columns
//   C (16x16): column n holds the 16 scores (identical across columns)
// f32 A layout: lane L holds A[L&15, 2*(L>>4)] , A[L&15, 2*(L>>4)+1]
// f32 B layout: lane L holds B[2*(L>>4), L&15], B[2*(L>>4)+1, L&15]
//   -> broadcast target chunk: same element for all 16 lanes of a half.
__global__ void __launch_bounds__(256)
sgns_score_kernel(const int* __restrict__ target,
                  const int* __restrict__ positive,
                  const int* __restrict__ negatives,
                  const float* __restrict__ target_emb,
                  const float* __restrict__ context_emb,
                  float* __restrict__ block_partials,
                  int Bn, int Kn, int Dn)
{
    __shared__ float wave_partials[8];
    const int lane = threadIdx.x & 31;
    const int wave = threadIdx.x >> 5;
    const int b    = blockIdx.x * 8 + wave;

    float local = 0.0f;
    if (b < Bn) {
        const int row  = lane & 15;   // A-matrix row fed by this lane
        const int half = lane >> 4;   // selects K-pair {0,1} vs {2,3}

        int cidx;
        if (row >= 1 && row <= Kn) cidx = negatives[b * Kn + (row - 1)];
        else                       cidx = positive[b];   // row 0 + dummy rows
        const int tidx = target[b];

        const float* aptr = context_emb + (size_t)cidx * (size_t)Dn + 2 * half;
        const float* bptr = target_emb  + (size_t)tidx * (size_t)Dn + 2 * half;

        v8f acc = {};
        const int chunks = Dn >> 2;   // D/4 k-chunks
        for (int c = 0; c < chunks; ++c) {
            v2f av = *(const v2f*)(aptr + 4 * c);
            v2f bv = *(const v2f*)(bptr + 4 * c);
            // 8 args: (neg_a, A, neg_b, B, c_mod, C, reuse_a, reuse_b)
            acc = __builtin_amdgcn_wmma_f32_16x16x4_f32(
                false, av, false, bv, (short)0, acc, false, false);
        }

        // Column 0 readers: lane 0 owns rows 0..7, lane 16 owns rows 8..15.
        if ((lane & 15) == 0) {
            const int mbase = half * 8;
            #pragma unroll
            for (int r = 0; r < 8; ++r) {
                const int m = mbase + r;
                float s = acc[r];
                if (m == 0)       local += softplus_fast(-s);  // positive pair
                else if (m <= Kn) local += softplus_fast(s);   // negatives
                // rows Kn+1..15: ignored (aliased duplicates)
            }
        }
    }

    // Wave reduction (only lanes 0 and 16 carry nonzero values).
    #pragma unroll
    for (int o = 16; o > 0; o >>= 1)
        local += __shfl_xor(local, o, 32);
    if (lane == 0) wave_partials[wave] = local;
    __syncthreads();
    if (threadIdx.x == 0) {
        float s = 0.0f;
        #pragma unroll
        for (int w = 0; w < 8; ++w) s += wave_partials[w];
        block_partials[blockIdx.x] = s;
    }
}

// Deterministic final reduction: sum block partials, scale by 1/B.
__global__ void __launch_bounds__(256)
sgns_reduce_kernel(const float* __restrict__ partials, int n,
                   float* __restrict__ out, float invB)
{
    __shared__ float sm[256];
    float s = 0.0f;
    for (int i = threadIdx.x; i < n; i += 256) s += partials[i];
    sm[threadIdx.x] = s;
    __syncthreads();
    for (int o = 128; o > 0; o >>= 1) {
        if (threadIdx.x < o) sm[threadIdx.x] += sm[threadIdx.x + o];
        __syncthreads();
    }
    if (threadIdx.x == 0) out[0] = sm[0] * invB;
}

extern "C" void kernel_launch(void* const* d_in, const int* in_sizes, int n_in,
                              void* d_out, int out_size, void* d_ws, size_t ws_size,
                              hipStream_t stream) {
    const int*   target    = (const int*)d_in[0];
    const int*   positive  = (const int*)d_in[1];
    const int*   negatives = (const int*)d_in[2];
    const float* temb      = (const float*)d_in[3];
    const float* cemb      = (const float*)d_in[4];
    float*       out       = (float*)d_out;

    const int Bn = in_sizes[0];            // 16384
    const int Kn = in_sizes[2] / Bn;       // 10
    const int Dn = 128;                    // embedding dim (fixed in reference)

    float* partials = (float*)d_ws;        // one float per block
    const int nblocks = (Bn + 7) / 8;      // 8 waves (samples) per 256-thread block

    sgns_score_kernel<<<nblocks, 256, 0, stream>>>(
        target, positive, negatives, temb, cemb, partials, Bn, Kn, Dn);
    sgns_reduce_kernel<<<1, 256, 0, stream>>>(
        partials, nblocks, out, 1.0f / (float)Bn);
}